// SpinalCPG_14276471292434
// MI455X (gfx1250) — compile-verified
//
// SpinalCPG on MI455X (gfx1250): one wave32 per batch element, one lane per
// neuron. Serial scan over T with an LDS ring buffer filled by
// global_load_async_to_lds_b128 (ASYNCcnt pipeline, depth 4 chunks x 4 steps),
// cross-lane group means via __shfl_xor / ds_bpermute, lane0 stores float4.

#include <hip/hip_runtime.h>

#define NW    8   // waves per block (blockDim = NW*32 = 256)
#define CH    4   // timesteps per async chunk (wave fetches CH*128B per op)
#define SLOTS 4   // ring depth in chunks = async pipeline depth

#if defined(__HIP_DEVICE_COMPILE__)
#  if __has_builtin(__builtin_amdgcn_global_load_async_to_lds_b128)
#    define HAVE_ASYNC 1
#  endif
#endif

#if defined(HAVE_ASYNC)
#  if __has_builtin(__builtin_amdgcn_s_wait_asynccnt)
#    define WAIT_ASYNC(n) __builtin_amdgcn_s_wait_asynccnt(n)
#  else
#    define WAIT_ASYNC(n) asm volatile("s_wait_asynccnt %0" ::"i"(n) : "memory")
#  endif

typedef int v4i __attribute__((ext_vector_type(4)));
typedef __attribute__((address_space(1))) v4i GV4;  // global int4
typedef __attribute__((address_space(3))) v4i LV4;  // LDS int4

// One async op per wave: each lane moves 16B.  Lane L fetches step t0+(L>>3),
// floats (L&7)*4 .. +3, into the matching LDS ring location.
static __device__ __forceinline__ void issue_async_b128(const float* g,
                                                        const float* l) {
  // AS(3) pointer value == LDS byte offset == low 32 bits of the flat address
  // (LDS aperture has zero low bits), so an integer round-trip is safe.
  __builtin_amdgcn_global_load_async_to_lds_b128(
      (GV4*)(unsigned long long)g,
      (LV4*)(unsigned)(unsigned long long)l,
      /*offset=*/0, /*cpol=*/0);
}
#endif

// One CPG timestep for neuron `lane` of one batch element (whole wave).
static __device__ __forceinline__ void cpg_step(
    float eps, float d, float tb, float& v, float& syn, float coef,
    int srcLane, bool isInh, bool extOn, bool isL, int lane, float* op) {
  // per-side drives
  const float dL = fmaf(fmaxf(tb, 0.0f), 0.3f, d);
  const float dR = fmaf(fmaxf(-tb, 0.0f), 0.3f, d);

  // group sums of syn: aligned quad (s4) and octet (s8) butterflies
  float a  = syn + __shfl_xor(syn, 1, 32);
  float s4 = a + __shfl_xor(a, 2, 32);
  float s8 = s4 + __shfl_xor(s4, 4, 32);
  // each lane gathers the one group-sum it needs (inh groups export s4,
  // exc groups export s8); coef folds weight / group-size.
  float merged = isInh ? s4 : s8;
  float gath   = __shfl(merged, srcLane, 32);

  const float Iext = extOn ? (isL ? dL : dR) : 0.0f;
  float u  = fmaf(coef, gath, Iext);
  u        = fmaf(eps, 0.08f, u);                // + NOISE*eps
  float vn = fmaf(0.7f, v, 0.3f * u);            // TAU_M leak + input
  const bool spk = vn >= 0.5f;                   // V_THRESH
  v   = spk ? 0.0f : vn;                         // V_RESET
  syn = fmaf(0.8f, syn, spk ? 1.0f : 0.0f);      // TAU_SYN decay + spike

  // motor means over lanes 12..15 (L) and 28..31 (R)
  float m1 = syn + __shfl_xor(syn, 1, 32);
  float m4 = m1 + __shfl_xor(m1, 2, 32);
  float mL = __shfl(m4, 12, 32) * 0.25f;
  float mR = __shfl(m4, 28, 32) * 0.25f;
  if (lane == 0) {
    *reinterpret_cast<float4*>(op) =
        make_float4(mL, mR, (mL + mR) * 0.5f, (mR - mL) * 2.0f);
  }
}

__global__ __launch_bounds__(NW * 32) void spinal_cpg_kernel(
    const float* __restrict__ drive, const float* __restrict__ turnb,
    const float* __restrict__ noise, const float* __restrict__ v0,
    const float* __restrict__ syn0, float* __restrict__ out, int Bn, int T) {
  const int lane = threadIdx.x & 31;
  const int wib  = threadIdx.x >> 5;
  const int b    = blockIdx.x * NW + wib;

#if defined(HAVE_ASYNC)
  __shared__ float ring[NW][SLOTS][CH][32];  // 16 KB / block
#endif

  if (b >= Bn) return;

  // per-lane constants from the fixed 32-neuron layout
  // groups of 4 (g3 = (lane>>2)&3): 0,1=exc  2=inh  3=mot  (per 16-lane side)
  const int   g3      = (lane >> 2) & 3;
  const float coef    = (g3 < 2) ? -0.25f : ((g3 == 2) ? 0.075f : 0.1f);
  const int   srcLane = (((lane >> 3) + 3) & 3) << 3;  // 0..7->24,8..15->0,16..23->8,24..31->16
  const bool  isInh   = (g3 == 2);
  const bool  extOn   = ((lane & 8) == 0);  // exc lanes only
  const bool  isL     = lane < 16;

  float v   = v0[(size_t)b * 32 + lane];
  float syn = syn0[(size_t)b * 32 + lane];

  const float* dP = drive + b;  // + t*Bn per step
  const float* tP = turnb + b;

  const int nCh = T / CH;

#if defined(HAVE_ASYNC)
  if (nCh >= SLOTS) {
    const size_t gStride = (size_t)CH * Bn * 32;  // floats per chunk
    const float* gp =
        noise + ((size_t)(lane >> 3) * Bn + b) * 32 + (size_t)(lane & 7) * 4;

    // prologue: fill the pipeline
    for (int s = 0; s < SLOTS; ++s) {
      issue_async_b128(gp, &ring[wib][s][lane >> 3][(lane & 7) * 4]);
      gp += gStride;
    }

    for (int c = 0; c < nCh; ++c) {
      const int slot = c & (SLOTS - 1);

      // prefetch wave-uniform drive/turn one pipeline-depth ahead
      int tpf = c * CH + SLOTS * CH;
      if (tpf >= T) tpf = T - 1;
      __builtin_prefetch(dP + (size_t)tpf * Bn, 0, 0);
      __builtin_prefetch(tP + (size_t)tpf * Bn, 0, 0);

      // async loads complete in order: <=SLOTS-1 outstanding => chunk c landed
      WAIT_ASYNC(SLOTS - 1);

#pragma unroll
      for (int s = 0; s < CH; ++s) {
        const int   t   = c * CH + s;
        const float eps = ring[wib][slot][s][lane];
        const float d   = dP[(size_t)t * Bn];
        const float tb  = tP[(size_t)t * Bn];
        cpg_step(eps, d, tb, v, syn, coef, srcLane, isInh, extOn, isL, lane,
                 out + ((size_t)t * Bn + b) * 4);
      }

      // refill the slot we just drained; at the tail re-issue the last valid
      // chunk (never read again) so WAIT_ASYNC(SLOTS-1) stays correct.
      issue_async_b128(gp, &ring[wib][slot][lane >> 3][(lane & 7) * 4]);
      if (c + SLOTS + 1 < nCh) gp += gStride;
    }

    // remainder steps (T % CH) via direct loads
    const float* np = noise + (size_t)b * 32 + lane;
    for (int t = nCh * CH; t < T; ++t) {
      const float eps = np[(size_t)t * (size_t)Bn * 32];
      const float d   = dP[(size_t)t * Bn];
      const float tb  = tP[(size_t)t * Bn];
      cpg_step(eps, d, tb, v, syn, coef, srcLane, isInh, extOn, isL, lane,
               out + ((size_t)t * Bn + b) * 4);
    }
  } else
#endif
  {
    // fallback: direct global loads with 1-step register prefetch
    const float* np      = noise + (size_t)b * 32 + lane;
    const size_t nStride = (size_t)Bn * 32;
    float eps = (T > 0) ? np[0] : 0.0f;
    for (int t = 0; t < T; ++t) {
      const float epsN = (t + 1 < T) ? np[(size_t)(t + 1) * nStride] : 0.0f;
      const float d  = dP[(size_t)t * Bn];
      const float tb = tP[(size_t)t * Bn];
      cpg_step(eps, d, tb, v, syn, coef, srcLane, isInh, extOn, isL, lane,
               out + ((size_t)t * Bn + b) * 4);
      eps = epsN;
    }
  }
}

extern "C" void kernel_launch(void* const* d_in, const int* in_sizes, int n_in,
                              void* d_out, int out_size, void* d_ws,
                              size_t ws_size, hipStream_t stream) {
  (void)n_in; (void)out_size; (void)d_ws; (void)ws_size;
  const float* drive = (const float*)d_in[0];  // [T,B]
  const float* turnb = (const float*)d_in[1];  // [T,B]
  const float* noise = (const float*)d_in[2];  // [T,B,32]
  const float* v0    = (const float*)d_in[3];  // [B,32]
  const float* syn0  = (const float*)d_in[4];  // [B,32]
  float* out = (float*)d_out;                  // [T,B,4]

  const int Bn = in_sizes[3] / 32;
  const int T  = (Bn > 0) ? in_sizes[0] / Bn : 0;
  if (Bn <= 0 || T <= 0) return;

  const int blocks = (Bn + NW - 1) / NW;
  spinal_cpg_kernel<<<blocks, NW * 32, 0, stream>>>(drive, turnb, noise, v0,
                                                    syn0, out, Bn, T);
}